// SSFL_56813827391742
// MI455X (gfx1250) — compile-verified
//
#include <hip/hip_runtime.h>
#include <hip/hip_bf16.h>

// ---- problem constants ----
#define NB   64     // batch
#define SS   64     // frames
#define CC   128    // in planes
#define PP   32     // parts
#define KK   1000   // classes
#define HPG  24     // hidden per group
#define C3   384    // 3*C
#define MM   4096   // N*S rows per part
#define YC   160    // padded cols in Y: [0,128)=feat, [128,152)=h, [152,160)=pad

// GEMM1 tiling
#define MB1   64    // rows per block
#define KCH   64    // k chunk staged in LDS
#define ASTR  386   // padded LDS row stride (dwords) for A panel (384 -> +2)
#define SM1_BYTES ((MB1 * ASTR + KCH * YC) * 4)   // 98816 + 40960 = 139776 B

typedef __attribute__((ext_vector_type(2))) float v2f;
typedef __attribute__((ext_vector_type(8))) float v8f;

// ---------------------------------------------------------------------------
// Pack x: (n,s,c,p) x3 tensors -> Xp[p][m=n*S+s][384] row-major contiguous.
// grid=(4096 m, 3 tensor), block=256. Reads lane-over-p (1 line/instr),
// writes contiguous.
// ---------------------------------------------------------------------------
__global__ __launch_bounds__(256) void k_pack_x(
        const float* __restrict__ tf, const float* __restrict__ ts,
        const float* __restrict__ tl, float* __restrict__ Xp) {
    int m = blockIdx.x;
    int tsel = blockIdx.y;
    const float* src = (tsel == 0) ? tf : ((tsel == 1) ? ts : tl);
    __shared__ float T[CC][PP + 1];               // 128 x 33, pad kills conflicts
    int t = threadIdx.x;
    int pl = t & 31, cq = t >> 5;                 // 8 c-rows per pass
    for (int j = 0; j < CC; j += 8) {
        int c = cq + j;
        T[c][pl] = src[((size_t)m * CC + c) * PP + pl];
    }
    __syncthreads();
    int il = t & 31, pw = t >> 5;
    for (int pj = 0; pj < PP; pj += 8) {
        int p = pw + pj;
        float* dst = Xp + ((size_t)p * MM + m) * C3 + (size_t)tsel * CC;
        for (int c0 = 0; c0 < CC; c0 += 32)
            dst[c0 + il] = T[c0 + il][p];
    }
}

// ---------------------------------------------------------------------------
// Pack weights k-major: Bt[p][k][col], col<128 -> wd[p,col,k],
// col in [128,152) -> w1[p,col-128,k], else 0.  grid=(32,384), block=160.
// ---------------------------------------------------------------------------
__global__ __launch_bounds__(160) void k_pack_b(
        const float* __restrict__ wd, const float* __restrict__ w1,
        float* __restrict__ Bt) {
    int p = blockIdx.x, k = blockIdx.y;
    int col = threadIdx.x;
    float v;
    if (col < CC)            v = wd[((size_t)p * CC + col) * C3 + k];
    else if (col < CC + HPG) v = w1[((size_t)p * HPG + (col - CC)) * C3 + k];
    else                     v = 0.0f;
    Bt[((size_t)p * C3 + k) * YC + col] = v;
}

// ---------------------------------------------------------------------------
// Phase 1 GEMM: Y[p][m][0:160] = Xp_p[64-row panel] * Bt_p, WMMA f32 16x16x4.
// grid=(64 row-panels, 32 parts), block=320 (10 waves, one 16-col tile each,
// 4 row-subtiles => 4 accumulators). A panel + B chunk staged in dynamic LDS.
// ---------------------------------------------------------------------------
__global__ __launch_bounds__(320) void k_gemm1(
        const float* __restrict__ Xp, const float* __restrict__ Bt,
        float* __restrict__ Y) {
    extern __shared__ float smem[];
    float* As = smem;                  // MB1 x ASTR  (padded rows)
    float* Bs = smem + MB1 * ASTR;     // KCH x YC

    int mb = blockIdx.x;
    int p  = blockIdx.y;
    int tid = threadIdx.x;
    int wave = tid >> 5, lane = tid & 31;
    int colq = lane & 15, kh = lane >> 4;
    int colg = wave * 16 + colq;       // output column 0..159

    // Stage A panel once: 64 rows x 384 (contiguous block in Xp), float2 copies.
    const float* ga = Xp + ((size_t)p * MM + (size_t)mb * MB1) * C3;
    for (int i2 = tid; i2 < MB1 * (C3 / 2); i2 += 320) {
        int r  = i2 / (C3 / 2);
        int c2 = i2 - r * (C3 / 2);
        float2 v = ((const float2*)ga)[(size_t)r * (C3 / 2) + c2];
        *(float2*)&As[r * ASTR + 2 * c2] = v;
    }

    v8f acc0 = {}, acc1 = {}, acc2 = {}, acc3 = {};
    const float* gb = Bt + (size_t)p * C3 * YC;

    for (int kc = 0; kc < C3; kc += KCH) {
        __syncthreads();               // A ready / previous B chunk consumed
        const float4* src = (const float4*)(gb + (size_t)kc * YC);
        for (int i4 = tid; i4 < (KCH * YC) / 4; i4 += 320)
            ((float4*)Bs)[i4] = src[i4];
        __syncthreads();

        for (int k0 = 0; k0 < KCH; k0 += 4) {
            int k  = k0 + 2 * kh;      // per-lane K pair per ISA A/B layout
            int ka = kc + k;
            v2f b;
            b[0] = Bs[k * YC + colg];
            b[1] = Bs[(k + 1) * YC + colg];
            v2f a0 = *(const v2f*)&As[(0 * 16 + colq) * ASTR + ka];
            v2f a1 = *(const v2f*)&As[(1 * 16 + colq) * ASTR + ka];
            v2f a2 = *(const v2f*)&As[(2 * 16 + colq) * ASTR + ka];
            v2f a3 = *(const v2f*)&As[(3 * 16 + colq) * ASTR + ka];
            acc0 = __builtin_amdgcn_wmma_f32_16x16x4_f32(false, a0, false, b, (short)0, acc0, false, false);
            acc1 = __builtin_amdgcn_wmma_f32_16x16x4_f32(false, a1, false, b, (short)0, acc1, false, false);
            acc2 = __builtin_amdgcn_wmma_f32_16x16x4_f32(false, a2, false, b, (short)0, acc2, false, false);
            acc3 = __builtin_amdgcn_wmma_f32_16x16x4_f32(false, a3, false, b, (short)0, acc3, false, false);
        }
    }

    // D layout: VGPR j -> rows (j, j+8) by lane half, col = lane&15
    float* yb = Y + ((size_t)p * MM + (size_t)mb * MB1) * YC;
    for (int j = 0; j < 8; ++j) {
        int r = j + 8 * kh;
        yb[(size_t)(0 * 16 + r) * YC + colg] = acc0[j];
        yb[(size_t)(1 * 16 + r) * YC + colg] = acc1[j];
        yb[(size_t)(2 * 16 + r) * YC + colg] = acc2[j];
        yb[(size_t)(3 * 16 + r) * YC + colg] = acc3[j];
    }
}

// ---------------------------------------------------------------------------
// Phase 2a: BN1 batch stats over (n,s) per (p,o). grid=(32,24), block=128
// ---------------------------------------------------------------------------
__global__ __launch_bounds__(128) void k_stats1(
        const float* __restrict__ Y, float* __restrict__ mu1,
        float* __restrict__ rstd1) {
    int p = blockIdx.x, o = blockIdx.y, t = threadIdx.x;
    __shared__ float ssum[128], ssq[128];
    const float* base = Y + (size_t)p * MM * YC + CC + o;
    float s = 0.f, q = 0.f;
    for (int j = t; j < MM; j += 128) {
        float v = base[(size_t)j * YC];
        s += v; q += v * v;
    }
    ssum[t] = s; ssq[t] = q;
    __syncthreads();
    for (int off = 64; off >= 1; off >>= 1) {
        if (t < off) { ssum[t] += ssum[t + off]; ssq[t] += ssq[t + off]; }
        __syncthreads();
    }
    if (t == 0) {
        float mu  = ssum[0] * (1.0f / MM);
        float var = ssq[0] * (1.0f / MM) - mu * mu;
        mu1[p * HPG + o]   = mu;
        rstd1[p * HPG + o] = rsqrtf(var + 1e-5f);
    }
}

// ---------------------------------------------------------------------------
// Phase 2b: per (n,p): score(s) -> sum/argmax -> wpv + selected gather.
// grid=(64 n, 32 p), block = 64 threads (t == s)
// ---------------------------------------------------------------------------
__global__ __launch_bounds__(64) void k_score_wpv(
        const float* __restrict__ Y, const float* __restrict__ tf,
        const float* __restrict__ mu1, const float* __restrict__ rstd1,
        const float* __restrict__ g1, const float* __restrict__ b1,
        const float* __restrict__ w2,
        float* __restrict__ out_wpv, float* __restrict__ out_sel) {
    int n = blockIdx.x, p = blockIdx.y, t = threadIdx.x;    // t == s
    __shared__ float sc[64], rv[64], rsum[64];
    __shared__ int   ri[64];

    const float* yrow = Y + ((size_t)p * MM + (size_t)n * SS + t) * YC;
    float acc = 0.f;
    for (int o = 0; o < HPG; ++o) {
        int po = p * HPG + o;
        float h = yrow[CC + o];
        h = (h - mu1[po]) * rstd1[po] * g1[po] + b1[po];
        h = (h >= 0.f) ? h : 0.01f * h;          // LeakyReLU(0.01)
        acc += h * w2[po];                        // w2 shape (P,1,HPG)
    }
    float score = 1.f / (1.f + __expf(-acc));
    sc[t] = score; rv[t] = score; ri[t] = t; rsum[t] = score;
    __syncthreads();
    for (int off = 32; off >= 1; off >>= 1) {
        if (t < off) {
            rsum[t] += rsum[t + off];
            // first-max semantics: strictly greater wins; ties keep lower idx
            if (rv[t + off] > rv[t] ||
                (rv[t + off] == rv[t] && ri[t + off] < ri[t])) {
                rv[t] = rv[t + off]; ri[t] = ri[t + off];
            }
        }
        __syncthreads();
    }
    float inv_sum = 1.f / rsum[0];
    int   smax    = ri[0];

    for (int cc = 0; cc < 2; ++cc) {
        int c = t + cc * 64;
        const float* ycol = Y + ((size_t)p * MM + (size_t)n * SS) * YC + c;
        float w = 0.f;
        for (int s = 0; s < SS; ++s)
            w += ycol[(size_t)s * YC] * sc[s];
        w *= inv_sum;
        size_t oidx = ((size_t)p * NB + n) * CC + c;
        out_wpv[oidx] = w;
        out_sel[oidx] = tf[(((size_t)n * SS + smax) * CC + c) * PP + p];
    }
}

// ---------------------------------------------------------------------------
// Phase 2c: per-part batch BN of wpv -> normalized part_feature. grid=32,blk=128
// ---------------------------------------------------------------------------
__global__ __launch_bounds__(128) void k_bn2_pf(
        const float* __restrict__ wpv, const float* __restrict__ g2,
        const float* __restrict__ b2, float* __restrict__ pf) {
    int p = blockIdx.x, c = threadIdx.x;
    const float* base = wpv + (size_t)p * NB * CC + c;
    float s = 0.f, q = 0.f;
    for (int n = 0; n < NB; ++n) {
        float v = base[(size_t)n * CC];
        s += v; q += v * v;
    }
    float mu  = s * (1.0f / NB);
    float var = q * (1.0f / NB) - mu * mu;
    float rs  = rsqrtf(var + 1e-5f);
    float ga  = g2[p * CC + c], be = b2[p * CC + c];
    for (int n = 0; n < NB; ++n) {
        float v = base[(size_t)n * CC];
        pf[((size_t)p * NB + n) * CC + c] = (v - mu) * rs * ga + be;
    }
}

// ---------------------------------------------------------------------------
// Phase 3: part_classification = pf[2048x128] x fc[128x1000] via WMMA f32.
// grid=(128 m-tiles, 8 col-groups), block=256 (8 waves, one 16-col tile each).
// Shared 16x128 A panel staged in LDS (padded stride 130).
// ---------------------------------------------------------------------------
__global__ __launch_bounds__(256) void k_fc(
        const float* __restrict__ pf, const float* __restrict__ fc,
        float* __restrict__ out) {
    __shared__ float As[16 * 130];
    int mt = blockIdx.x, ng = blockIdx.y;
    int tid = threadIdx.x;
    int wave = tid >> 5, lane = tid & 31;
    int colq = lane & 15, kh = lane >> 4;
    int nt  = ng * 8 + wave;            // 0..63 column tiles (cols 0..1023)
    int col = nt * 16 + colq;
    float cmask = (col < KK) ? 1.0f : 0.0f;
    int colc = (col < KK) ? col : 0;

    const float* ga = pf + (size_t)mt * 16 * CC;   // contiguous 16x128 panel
    for (int i2 = tid; i2 < 16 * (CC / 2); i2 += 256) {
        int r  = i2 / (CC / 2);
        int c2 = i2 - r * (CC / 2);
        *(float2*)&As[r * 130 + 2 * c2] = ((const float2*)ga)[r * (CC / 2) + c2];
    }
    __syncthreads();

    v8f acc = {};
    for (int k0 = 0; k0 < CC; k0 += 4) {
        int k = k0 + 2 * kh;
        v2f a = *(const v2f*)&As[colq * 130 + k];
        v2f b;
        b[0] = fc[(size_t)k * KK + colc] * cmask;
        b[1] = fc[(size_t)(k + 1) * KK + colc] * cmask;
        acc = __builtin_amdgcn_wmma_f32_16x16x4_f32(
                  false, a, false, b, (short)0, acc, false, false);
    }
    if (col < KK) {
        for (int j = 0; j < 8; ++j) {
            int r = mt * 16 + j + 8 * kh;
            out[(size_t)r * KK + col] = acc[j];
        }
    }
}

// ---------------------------------------------------------------------------
extern "C" void kernel_launch(void* const* d_in, const int* in_sizes, int n_in,
                              void* d_out, int out_size, void* d_ws, size_t ws_size,
                              hipStream_t stream) {
    const float* tf = (const float*)d_in[0];
    const float* ts = (const float*)d_in[1];
    const float* tl = (const float*)d_in[2];
    const float* w1 = (const float*)d_in[3];
    const float* g1 = (const float*)d_in[4];
    const float* b1 = (const float*)d_in[5];
    const float* w2 = (const float*)d_in[6];
    const float* wd = (const float*)d_in[7];
    const float* g2 = (const float*)d_in[8];
    const float* b2 = (const float*)d_in[9];
    const float* fc = (const float*)d_in[10];

    float* out = (float*)d_out;
    float* out_pc  = out;                              // (P,N,K)  2,048,000
    float* out_wpv = out + (size_t)PP * NB * KK;       // (P,N,C)    262,144
    float* out_sel = out_wpv + (size_t)PP * NB * CC;   // (P,N,C)    262,144

    float* ws    = (float*)d_ws;
    float* Xp    = ws;                                   // 50,331,648
    float* Bt    = Xp + (size_t)PP * MM * C3;            //  1,966,080
    float* Y     = Bt + (size_t)PP * C3 * YC;            // 20,971,520
    float* mu1   = Y + (size_t)PP * MM * YC;             //        768
    float* rstd1 = mu1 + PP * HPG;                       //        768
    float* pf    = rstd1 + PP * HPG;                     //    262,144

    k_pack_x  <<<dim3(MM, 3),        256, 0, stream>>>(tf, ts, tl, Xp);
    k_pack_b  <<<dim3(PP, C3),       160, 0, stream>>>(wd, w1, Bt);
    k_gemm1   <<<dim3(MM / MB1, PP), 320, SM1_BYTES, stream>>>(Xp, Bt, Y);
    k_stats1  <<<dim3(PP, HPG),      128, 0, stream>>>(Y, mu1, rstd1);
    k_score_wpv<<<dim3(NB, PP),       64, 0, stream>>>(Y, tf, mu1, rstd1, g1, b1,
                                                       w2, out_wpv, out_sel);
    k_bn2_pf  <<<PP,                 128, 0, stream>>>(out_wpv, g2, b2, pf);
    k_fc      <<<dim3((PP * NB) / 16, 8), 256, 0, stream>>>(pf, fc, out_pc);
}